// Batch_Loss_8804682957501
// MI455X (gfx1250) — compile-verified
//
#include <hip/hip_runtime.h>
#include <hip/hip_bf16.h>

// MI455X / gfx1250: wave32, WMMA (not MFMA).
// Workload is HBM-bound batched row reductions (B=4096 rows, N=8192 cols, f32).
// Pass1: dot[b], neg[b]. Pass2: comm[b] (needs dot[b-1] -> separate kernel).
// Pass3: single-block finalize (SR, loss, prod, MDD scan).

#define B_ROWS 4096
#define N_COLS 8192
#define COMMISSION_RATIO 0.0025f
#define INTEREST_RATE 8e-05f

typedef __attribute__((ext_vector_type(2))) float v2f;
typedef __attribute__((ext_vector_type(8))) float v8f;

// XOR-butterfly lane exchange via ds_swizzle_b32 (group-of-32 mode):
// offset[14:10]=xor_mask, offset[9:5]=or_mask(0), offset[4:0]=and_mask(0x1F).
// Single DS op, no shuffle-width clamp boilerplate. All lanes active here.
#define SWZ_XOR(x, m) __int_as_float(__builtin_amdgcn_ds_swizzle( \
    __float_as_int(x), ((m) << 10) | 0x1F))

// Wave32 sum of one f32 per lane using V_WMMA_F32_16X16X4_F32.
// A = ones(16x4), B: b.x = partial, b.y = 0.
// D[m][n] = sum_k B[k][n] = p[lane n] + p[lane n+16]  (replicated over rows m).
// c[0] in lane L then holds colsum[L % 16]; 4 xor-swizzles finish the 16-way sum.
// Requires EXEC == all ones (call from uniform full-wave control flow only).
__device__ __forceinline__ float wave_sum32(float p) {
  v2f a; a.x = 1.0f; a.y = 1.0f;       // A-matrix: all ones
  v2f b; b.x = p;    b.y = 0.0f;       // B-matrix: partials in rows {0,2}
  v8f c = {};
  c = __builtin_amdgcn_wmma_f32_16x16x4_f32(
      /*neg_a=*/false, a, /*neg_b=*/false, b,
      /*c_mod=*/(short)0, c, /*reuse_a=*/false, /*reuse_b=*/false);
  float s = c[0];
  s += SWZ_XOR(s, 1);
  s += SWZ_XOR(s, 2);
  s += SWZ_XOR(s, 4);
  s += SWZ_XOR(s, 8);
  return s;  // total of all 32 lanes, in every lane
}

// Block (256 threads = 8 wave32) sum. Safe for repeated calls (trailing barrier).
__device__ __forceinline__ float block_sum256(float v, float* scratch) {
  const int lane = threadIdx.x & 31;
  const int wid  = threadIdx.x >> 5;
  v = wave_sum32(v);
  if (lane == 0) scratch[wid] = v;
  __syncthreads();
  if (wid == 0) {
    float t = (lane < 8) ? scratch[lane] : 0.0f;
    t = wave_sum32(t);                 // wave 0 fully active: EXEC all ones
    if (lane == 0) scratch[0] = t;
  }
  __syncthreads();
  float r = scratch[0];
  __syncthreads();                     // allow scratch reuse by next call
  return r;
}

// Pass 1: per-row dot product and negative-part sum. One block per row.
__global__ __launch_bounds__(256) void pass1_row_reduce(
    const float* __restrict__ w, const float* __restrict__ y,
    float* __restrict__ dotv, float* __restrict__ negv) {
  __shared__ float scr[8];
  const int b = blockIdx.x;
  const float4* wr = (const float4*)(w + (size_t)b * N_COLS);
  const float4* yr = (const float4*)(y + (size_t)b * N_COLS);
  float d = 0.0f, ng = 0.0f;
#pragma unroll
  for (int i = threadIdx.x; i < N_COLS / 4; i += 256) {
    float4 a = wr[i];
    float4 c = yr[i];
    float p0 = a.x * c.x, p1 = a.y * c.y, p2 = a.z * c.z, p3 = a.w * c.w;
    d  += (p0 + p1) + (p2 + p3);
    ng += (fminf(p0, 0.0f) + fminf(p1, 0.0f)) +
          (fminf(p2, 0.0f) + fminf(p3, 0.0f));
  }
  float D  = block_sum256(d, scr);
  float NG = block_sum256(ng, scr);
  if (threadIdx.x == 0) {
    dotv[b] = D;
    negv[b] = NG;
  }
}

// Pass 2: commission term comm[b] = sum_n |w[b-1]*y[b-1]/dot[b-1] - w[b]|.
__global__ __launch_bounds__(256) void pass2_commission(
    const float* __restrict__ w, const float* __restrict__ y,
    const float* __restrict__ dotv, float* __restrict__ commv) {
  __shared__ float scr[8];
  const int b = blockIdx.x;
  if (b == 0) {
    if (threadIdx.x == 0) commv[0] = 0.0f;   // pure_pc[0] == 1.0
    return;
  }
  const float inv = 1.0f / dotv[b - 1];
  const float4* wp = (const float4*)(w + (size_t)(b - 1) * N_COLS);
  const float4* yp = (const float4*)(y + (size_t)(b - 1) * N_COLS);
  const float4* wc = (const float4*)(w + (size_t)b * N_COLS);
  float s = 0.0f;
#pragma unroll
  for (int i = threadIdx.x; i < N_COLS / 4; i += 256) {
    float4 a = wp[i];
    float4 c = yp[i];
    float4 u = wc[i];
    s += fabsf(fmaf(a.x * c.x, inv, -u.x)) + fabsf(fmaf(a.y * c.y, inv, -u.y)) +
         fabsf(fmaf(a.z * c.z, inv, -u.z)) + fabsf(fmaf(a.w * c.w, inv, -u.w));
  }
  float S = block_sum256(s, scr);
  if (threadIdx.x == 0) commv[b] = S;
}

// Pass 3: single-block finalize. reward[b] = dot[b]*pure_pc[b] + neg[b]*IR.
// Outputs: out[0]=loss, out[1]=portfolio_value, out[2]=SR, out[3]=MDD.
__global__ __launch_bounds__(256) void pass3_finalize(
    const float* __restrict__ dotv, const float* __restrict__ negv,
    const float* __restrict__ commv, float* __restrict__ out) {
  __shared__ float rbuf[B_ROWS];       // 16 KB of the 320 KB LDS
  __shared__ float scr[8];
  __shared__ float pscr[256];
  float s1 = 0.0f, s2 = 0.0f, sl = 0.0f, pr = 1.0f;
  for (int b = threadIdx.x; b < B_ROWS; b += 256) {
    float pc = (b == 0) ? 1.0f : 1.0f - commv[b] * COMMISSION_RATIO;
    float r  = dotv[b] * pc + negv[b] * INTEREST_RATE;
    rbuf[b] = r;
    float x = r - 1.0f;
    s1 += x;
    s2 += x * x;
    sl += logf(r);
    pr *= r;
  }
  float S1 = block_sum256(s1, scr);
  float S2 = block_sum256(s2, scr);
  float SL = block_sum256(sl, scr);
  pscr[threadIdx.x] = pr;
  __syncthreads();
  if (threadIdx.x == 0) {
    float P = 1.0f;
    for (int i = 0; i < 256; ++i) P *= pscr[i];           // product reduction
    // Sequential cumprod / cummax scan for max drawdown (4096 iters, trivial).
    float pv = 1.0f, rmax = 0.0f, mdd = 0.0f;
    for (int b = 0; b < B_ROWS; ++b) {
      pv *= rbuf[b];
      rmax = fmaxf(rmax, pv);
      mdd  = fmaxf(mdd, 1.0f - pv / rmax);
    }
    const float invB = 1.0f / (float)B_ROWS;
    float mean = S1 * invB;
    float var  = (S2 - S1 * S1 * invB) * (1.0f / (float)(B_ROWS - 1));
    out[0] = -SL * invB;               // loss = -mean(log(reward))
    out[1] = P;                        // portfolio_value = prod(reward)
    out[2] = mean / sqrtf(var);        // SR (ddof=1)
    out[3] = mdd;                      // MDD
  }
}

extern "C" void kernel_launch(void* const* d_in, const int* in_sizes, int n_in,
                              void* d_out, int out_size, void* d_ws, size_t ws_size,
                              hipStream_t stream) {
  const float* w = (const float*)d_in[0];  // (4096, 1, 8192) f32
  const float* y = (const float*)d_in[1];  // (4096, 8192, 1) f32 (same flat layout)
  float* out = (float*)d_out;              // 4 floats: loss, pv, SR, MDD

  float* ws   = (float*)d_ws;              // needs 3*4096 floats = 48 KB
  float* dotv = ws;
  float* negv = ws + B_ROWS;
  float* comm = ws + 2 * B_ROWS;

  pass1_row_reduce<<<B_ROWS, 256, 0, stream>>>(w, y, dotv, negv);
  pass2_commission<<<B_ROWS, 256, 0, stream>>>(w, y, dotv, comm);
  pass3_finalize<<<1, 256, 0, stream>>>(dotv, negv, comm, out);
}